// GATModel_74380243632482
// MI455X (gfx1250) — compile-verified
//
#include <hip/hip_runtime.h>
#include <hip/hip_bf16.h>
#include <stdint.h>

typedef __attribute__((ext_vector_type(2))) float v2f;
typedef __attribute__((ext_vector_type(8))) float v8f;

#define N_NODES 50000
#define N_EDGES 800000
#define N_GRAPHS 64
#define HID 512
#define E_TOT (N_EDGES + N_NODES)   // self-loops appended

// ---------------------------------------------------------------- utilities

__global__ __launch_bounds__(256) void fill_u32_kernel(uint32_t* __restrict__ p,
                                                       uint32_t v, int n) {
  int i = blockIdx.x * blockDim.x + threadIdx.x;
  int stride = gridDim.x * blockDim.x;
  for (; i < n; i += stride) p[i] = v;
}

// monotone order-preserving float<->uint map (for atomicMax on floats)
__device__ __forceinline__ uint32_t f2ord(float f) {
  uint32_t u = __float_as_uint(f);
  return (u & 0x80000000u) ? ~u : (u | 0x80000000u);
}
__device__ __forceinline__ float ord2f(uint32_t u) {
  u = (u & 0x80000000u) ? (u ^ 0x80000000u) : ~u;
  return __uint_as_float(u);
}

// ---------------------------------------------------------------- WMMA GEMM
// C[M,N] = A[M,K] * B[K,N], row-major fp32, via V_WMMA_F32_16X16X4_F32.
// One wave computes one 16x16 C tile. Requires M%16==0, N%16==0, K%4==0.
__global__ __launch_bounds__(128) void gemm_f32_wmma(const float* __restrict__ A,
                                                     const float* __restrict__ B,
                                                     float* __restrict__ C,
                                                     int M, int K, int N) {
  const int lane = threadIdx.x & 31;
  const int wave = blockIdx.x * (blockDim.x >> 5) + (threadIdx.x >> 5);
  const int tilesN = N >> 4;
  const int tm = wave / tilesN;
  const int tn = wave - tm * tilesN;
  if (tm >= (M >> 4)) return;            // wave-uniform: EXEC all-1 for WMMA

  const int r  = lane & 15;              // row-in-tile (A) / col-in-tile (B,C)
  const int kh = (lane >> 4) << 1;       // 0 or 2: K half handled by this lane

  const float* Ap = A + (size_t)(tm * 16 + r) * K + kh;
  const float* Bp = B + (size_t)kh * N + (tn * 16 + r);

  v8f acc = {0.f, 0.f, 0.f, 0.f, 0.f, 0.f, 0.f, 0.f};
  for (int k0 = 0; k0 < K; k0 += 4) {
    v2f a, b;
    a.x = Ap[0];
    a.y = Ap[1];
    b.x = Bp[0];
    b.y = Bp[(size_t)N];
    acc = __builtin_amdgcn_wmma_f32_16x16x4_f32(
        /*neg_a=*/false, a, /*neg_b=*/false, b,
        /*c_mod=*/(short)0, acc, /*reuse_a=*/false, /*reuse_b=*/false);
    Ap += 4;
    Bp += (size_t)4 * N;
  }

  const int row0 = tm * 16 + ((lane >> 4) << 3);   // M = j + 8*(lane>=16)
  float* Cp = C + (size_t)row0 * N + (tn * 16 + r);
#pragma unroll
  for (int j = 0; j < 8; ++j) Cp[(size_t)j * N] = acc[j];
}

// ------------------------------------------------------- per-node att dots
// asrc[n] = h[n,:].att[0,:]   adst[n] = h[n,:].att[1,:]   (wave per node)
__global__ __launch_bounds__(256) void node_att_kernel(const float* __restrict__ H,
                                                       const float* __restrict__ att,
                                                       float* __restrict__ asrc,
                                                       float* __restrict__ adst,
                                                       int D) {
  int node = blockIdx.x * (blockDim.x >> 5) + (threadIdx.x >> 5);
  int lane = threadIdx.x & 31;
  if (node >= N_NODES) return;
  float s0 = 0.f, s1 = 0.f;
  for (int d = lane; d < D; d += 32) {
    float hv = H[(size_t)node * D + d];
    s0 += hv * att[d];
    s1 += hv * att[D + d];
  }
#pragma unroll
  for (int off = 16; off > 0; off >>= 1) {
    s0 += __shfl_down(s0, off, 32);
    s1 += __shfl_down(s1, off, 32);
  }
  if (lane == 0) { asrc[node] = s0; adst[node] = s1; }
}

// ------------------------------------------------------------- edge kernels

__global__ __launch_bounds__(256) void edge_logit_max_kernel(
    const int* __restrict__ esrc, const int* __restrict__ edst,
    const float* __restrict__ asrc, const float* __restrict__ adst,
    float* __restrict__ logit, uint32_t* __restrict__ menc) {
  int e = blockIdx.x * blockDim.x + threadIdx.x;
  int stride = gridDim.x * blockDim.x;
  for (; e < E_TOT; e += stride) {
    int s = (e < N_EDGES) ? esrc[e] : (e - N_EDGES);
    int d = (e < N_EDGES) ? edst[e] : (e - N_EDGES);
    float l = asrc[s] + adst[d];
    l = (l > 0.f) ? l : 0.2f * l;               // leaky_relu slope 0.2
    logit[e] = l;
    atomicMax(&menc[d], f2ord(l));
  }
}

__global__ __launch_bounds__(256) void edge_exp_sum_kernel(
    const int* __restrict__ edst, float* __restrict__ logit /* -> expw */,
    const uint32_t* __restrict__ menc, float* __restrict__ ssum) {
  int e = blockIdx.x * blockDim.x + threadIdx.x;
  int stride = gridDim.x * blockDim.x;
  for (; e < E_TOT; e += stride) {
    int d = (e < N_EDGES) ? edst[e] : (e - N_EDGES);
    float w = __expf(logit[e] - ord2f(menc[d]));
    logit[e] = w;
    unsafeAtomicAdd(&ssum[d], w);
  }
}

// wave per edge: out[dst,:] += (expw[e]/ssum[dst]) * h[src,:]
__global__ __launch_bounds__(256) void edge_agg_kernel(
    const int* __restrict__ esrc, const int* __restrict__ edst,
    const float* __restrict__ expw, const float* __restrict__ ssum,
    const float* __restrict__ H, float* __restrict__ O, int D) {
  int e = blockIdx.x * (blockDim.x >> 5) + (threadIdx.x >> 5);
  if (e >= E_TOT) return;
  int lane = threadIdx.x & 31;
  int s = (e < N_EDGES) ? esrc[e] : (e - N_EDGES);
  int d = (e < N_EDGES) ? edst[e] : (e - N_EDGES);
  float coef = expw[e] / ssum[d];
  const float* hrow = H + (size_t)s * D;
  float* orow = O + (size_t)d * D;
  for (int f = lane; f < D; f += 32) unsafeAtomicAdd(&orow[f], coef * hrow[f]);
}

__global__ __launch_bounds__(256) void bias_relu_kernel(float* __restrict__ X,
                                                        const float* __restrict__ b,
                                                        int n, int D) {
  int i = blockIdx.x * blockDim.x + threadIdx.x;
  int stride = gridDim.x * blockDim.x;
  for (; i < n; i += stride) {
    float v = X[i] + b[i % D];
    X[i] = (v > 0.f) ? v : 0.f;
  }
}

// ---------------------------------------------------------------- pooling

__global__ __launch_bounds__(256) void pool_sum_kernel(const float* __restrict__ X,
                                                       const int* __restrict__ batch,
                                                       float* __restrict__ sums,
                                                       uint32_t* __restrict__ cnt,
                                                       int D) {
  int node = blockIdx.x * (blockDim.x >> 5) + (threadIdx.x >> 5);
  int lane = threadIdx.x & 31;
  if (node >= N_NODES) return;
  int g = batch[node];
  for (int d = lane; d < D; d += 32)
    unsafeAtomicAdd(&sums[(size_t)g * D + d], X[(size_t)node * D + d]);
  if (lane == 0) atomicAdd(&cnt[g], 1u);
}

__global__ __launch_bounds__(256) void pool_div_kernel(const float* __restrict__ sums,
                                                       const uint32_t* __restrict__ cnt,
                                                       float* __restrict__ gc, int D) {
  int i = blockIdx.x * blockDim.x + threadIdx.x;
  if (i < N_GRAPHS * D) gc[i] = sums[i] / fmaxf((float)cnt[i / D], 1.0f);
}

// final fc2: out[g] = sum_k fc1o[g,k]*w[k] + b   (wave per graph)
__global__ __launch_bounds__(256) void fc2_kernel(const float* __restrict__ X,
                                                  const float* __restrict__ w,
                                                  const float* __restrict__ b,
                                                  float* __restrict__ out) {
  int g = blockIdx.x * (blockDim.x >> 5) + (threadIdx.x >> 5);
  int lane = threadIdx.x & 31;
  if (g >= N_GRAPHS) return;
  float s = 0.f;
  for (int k = lane; k < HID; k += 32) s += X[(size_t)g * HID + k] * w[k];
#pragma unroll
  for (int off = 16; off > 0; off >>= 1) s += __shfl_down(s, off, 32);
  if (lane == 0) out[g] = s + b[0];
}

// ---------------------------------------------------------------- launcher

extern "C" void kernel_launch(void* const* d_in, const int* in_sizes, int n_in,
                              void* d_out, int out_size, void* d_ws, size_t ws_size,
                              hipStream_t stream) {
  const float* feature = (const float*)d_in[0];
  const int*   eidx    = (const int*)d_in[1];
  const int*   batch   = (const int*)d_in[2];
  const float* W1   = (const float*)d_in[3];
  const float* att1 = (const float*)d_in[4];
  const float* b1   = (const float*)d_in[5];
  const float* W2   = (const float*)d_in[6];
  const float* att2 = (const float*)d_in[7];
  const float* b2   = (const float*)d_in[8];
  const float* W3   = (const float*)d_in[9];
  const float* att3 = (const float*)d_in[10];
  const float* b3   = (const float*)d_in[11];
  const float* fc1w = (const float*)d_in[12];
  const float* fc1b = (const float*)d_in[13];
  const float* fc2w = (const float*)d_in[14];
  const float* fc2b = (const float*)d_in[15];

  const int* esrc = eidx;
  const int* edst = eidx + N_EDGES;

  // workspace layout (fp32 words)
  float* ws = (float*)d_ws;
  size_t off = 0;
  float*    H    = ws + off; off += (size_t)N_NODES * 256;   // lin output
  float*    O    = ws + off; off += (size_t)N_NODES * 256;   // agg output / next x
  float*    Asrc = ws + off; off += N_NODES;
  float*    Adst = ws + off; off += N_NODES;
  uint32_t* Menc = (uint32_t*)(ws + off); off += N_NODES;
  float*    Ssum = ws + off; off += N_NODES;
  float*    Elog = ws + off; off += E_TOT;                   // logit -> exp weight
  float*    Sums = ws + off; off += (size_t)N_GRAPHS * 256;
  uint32_t* Cnt  = (uint32_t*)(ws + off); off += N_GRAPHS;
  float*    Gc   = ws + off; off += (size_t)N_GRAPHS * 256;
  float*    Fc1o = ws + off; off += (size_t)N_GRAPHS * HID;
  (void)ws_size; (void)n_in; (void)in_sizes; (void)out_size;

  auto layer = [&](const float* x, const float* W, const float* att,
                   const float* b, int Din, int Dout) {
    int tiles = (N_NODES / 16) * (Dout / 16);
    gemm_f32_wmma<<<(tiles + 3) / 4, 128, 0, stream>>>(x, W, H, N_NODES, Din, Dout);
    node_att_kernel<<<(N_NODES + 7) / 8, 256, 0, stream>>>(H, att, Asrc, Adst, Dout);
    fill_u32_kernel<<<512, 256, 0, stream>>>(Menc, 0u, N_NODES);
    fill_u32_kernel<<<512, 256, 0, stream>>>((uint32_t*)Ssum, 0u, N_NODES);
    fill_u32_kernel<<<4096, 256, 0, stream>>>((uint32_t*)O, 0u, N_NODES * Dout);
    edge_logit_max_kernel<<<2048, 256, 0, stream>>>(esrc, edst, Asrc, Adst, Elog, Menc);
    edge_exp_sum_kernel<<<2048, 256, 0, stream>>>(edst, Elog, Menc, Ssum);
    edge_agg_kernel<<<(E_TOT + 7) / 8, 256, 0, stream>>>(esrc, edst, Elog, Ssum, H, O, Dout);
    bias_relu_kernel<<<4096, 256, 0, stream>>>(O, b, N_NODES * Dout, Dout);
  };

  layer(feature, W1, att1, b1, 64, 64);    // x = input features
  layer(O,       W2, att2, b2, 64, 128);   // O read by GEMM, then rewritten
  layer(O,       W3, att3, b3, 128, 256);

  // global mean pool
  fill_u32_kernel<<<64, 256, 0, stream>>>((uint32_t*)Sums, 0u, N_GRAPHS * 256);
  fill_u32_kernel<<<1, 64, 0, stream>>>(Cnt, 0u, N_GRAPHS);
  pool_sum_kernel<<<(N_NODES + 7) / 8, 256, 0, stream>>>(O, batch, Sums, Cnt, 256);
  pool_div_kernel<<<(N_GRAPHS * 256 + 255) / 256, 256, 0, stream>>>(Sums, Cnt, Gc, 256);

  // fc1 (64x256 @ 256x512) via WMMA, + bias/relu, then fc2 -> d_out
  {
    int tiles = (N_GRAPHS / 16) * (HID / 16);
    gemm_f32_wmma<<<(tiles + 3) / 4, 128, 0, stream>>>(Gc, fc1w, Fc1o, N_GRAPHS, 256, HID);
    bias_relu_kernel<<<128, 256, 0, stream>>>(Fc1o, fc1b, N_GRAPHS * HID, HID);
    fc2_kernel<<<(N_GRAPHS + 7) / 8, 256, 0, stream>>>(Fc1o, fc2w, fc2b, (float*)d_out);
  }
}